// EDANLIFunction_7267084665401
// MI455X (gfx1250) — compile-verified
//
#include <hip/hip_runtime.h>
#include <hip/hip_fp16.h>
#include <stdint.h>

// ---------------------------------------------------------------------------
// EDA piecewise-linear LUT activation, specialized for MI455X (gfx1250).
//
// out = f32( T[ f16_bits( f16(x) ) ] ) for a 65536-entry fp16->fp16 table T
// that depends only on the per-launch config. 128KB table fits in CDNA5's
// 320KB LDS. Per-element cost collapses to cvt + ds_load_u16 + cvt, far under
// the 23.3 TB/s HBM roofline (~22us for the 512MB stream).
// ---------------------------------------------------------------------------

#define TABLE_ENTRIES 65536
#define TABLE_BYTES   (TABLE_ENTRIES * 2)

typedef float v4f __attribute__((ext_vector_type(4)));

// ---- Kernel A: build fp16->fp16 table, bit-exact vs reference math --------
__global__ void __launch_bounds__(256)
eda_build_table(const float* __restrict__ bin_starts,
                const float* __restrict__ bin_ends,
                const int*   __restrict__ base_offsets,
                const int*   __restrict__ k_bits,
                const float* __restrict__ lut_values,
                int n_bins, int lut_size,
                uint16_t* __restrict__ table) {
#pragma clang fp contract(off)
    unsigned p = blockIdx.x * blockDim.x + threadIdx.x;
    if (p >= TABLE_ENTRIES) return;

    float xf = __half2float(__ushort_as_half((unsigned short)p));

    // searchsorted(bin_starts, xf, side='left') == count(bin_starts < xf)
    int cnt = 0;
    for (int i = 0; i < n_bins; ++i) cnt += (bin_starts[i] < xf) ? 1 : 0;
    int bin = cnt - 1;
    if (bin < 0) bin = 0;
    if (bin > n_bins - 1) bin = n_bins - 1;

    float bs = bin_starts[bin];
    float be = bin_ends[bin];
    int   bo = base_offsets[bin];
    int   kb = k_bits[bin];

    float bw  = fmaxf(be - bs, 1e-30f);
    float rel = (xf - bs) / bw;
    rel = fminf(fmaxf(rel, 0.0f), 1.0f - 1e-7f);

    float nmic   = (float)(1 << kb);
    float scaled = rel * nmic;
    int   mi     = (int)scaled;
    int   mmax   = (int)(nmic - 1.0f);
    if (mi > mmax) mi = mmax;
    float t = scaled - (float)mi;
    t = fminf(fmaxf(t, 0.0f), 1.0f);

    int idx0 = bo + mi;
    if (idx0 < 0) idx0 = 0;
    if (idx0 > lut_size - 1) idx0 = lut_size - 1;
    int idx1 = bo + mi + 1;
    if (idx1 < 0) idx1 = 0;
    if (idx1 > lut_size - 1) idx1 = lut_size - 1;

    __half y0 = __float2half(lut_values[idx0]);
    __half y1 = __float2half(lut_values[idx1]);
    __half df = __hsub(y1, y0);                 // fp16 subtraction, as in torch

    float prod = t * __half2float(df);          // contract(off): two roundings,
    float yf   = __half2float(y0) + prod;       // matches reference exactly
    __half y   = __float2half(yf);

    if (xf < bin_starts[0])          y = __float2half(lut_values[0]);
    if (xf > bin_ends[n_bins - 1])   y = __float2half(lut_values[lut_size - 1]);

    table[p] = __half_as_ushort(y);
}

// ---- Kernel B: async-stage table to LDS, stream x through it --------------
__device__ __forceinline__ float eda_lookup(float v, const uint16_t* s_tab) {
    unsigned short b = __half_as_ushort(__float2half(v));
    return __half2float(__ushort_as_half(s_tab[b]));
}

__global__ void __launch_bounds__(256)
eda_apply(const float* __restrict__ x,
          float* __restrict__ y,
          const uint16_t* __restrict__ table,
          long long n) {
    __shared__ uint16_t s_tab[TABLE_ENTRIES];   // 128 KB: needs CDNA5 320KB LDS

    // Stage 128KB global -> LDS with CDNA5 async copies (no VGPR round-trip,
    // tracked by ASYNCcnt). LDS byte address = low 32 bits of generic address
    // (flat->LDS mapping truncates to addr[31:0], ISA 10.2).
    {
        const uint32_t lds_base = (uint32_t)(uintptr_t)(&s_tab[0]);
        for (int i = threadIdx.x; i < (int)(TABLE_BYTES / 16); i += blockDim.x) {
            uint32_t lds_addr = lds_base + 16u * (uint32_t)i;
            uint64_t gaddr    = (uint64_t)(uintptr_t)(table) + 16ull * (uint64_t)i;
            asm volatile("global_load_async_to_lds_b128 %0, %1, off"
                         :: "v"(lds_addr), "v"(gaddr)
                         : "memory");
        }
        asm volatile("s_wait_asynccnt 0x0" ::: "memory");
    }
    __syncthreads();

    const long long n4     = n >> 2;
    const long long stride = (long long)gridDim.x * blockDim.x;
    const v4f* __restrict__ x4 = (const v4f*)x;
    v4f* __restrict__       y4 = (v4f*)y;

    long long i = (long long)blockIdx.x * blockDim.x + threadIdx.x;

    // 4-deep unroll for memory-level parallelism; non-temporal policy keeps
    // the 512MB stream from evicting the L2-resident table.
    for (; i + 3 * stride < n4; i += 4 * stride) {
        v4f a = __builtin_nontemporal_load(x4 + i);
        v4f b = __builtin_nontemporal_load(x4 + i + stride);
        v4f c = __builtin_nontemporal_load(x4 + i + 2 * stride);
        v4f d = __builtin_nontemporal_load(x4 + i + 3 * stride);
        v4f ra, rb, rc, rd;
        ra.x = eda_lookup(a.x, s_tab); ra.y = eda_lookup(a.y, s_tab);
        ra.z = eda_lookup(a.z, s_tab); ra.w = eda_lookup(a.w, s_tab);
        rb.x = eda_lookup(b.x, s_tab); rb.y = eda_lookup(b.y, s_tab);
        rb.z = eda_lookup(b.z, s_tab); rb.w = eda_lookup(b.w, s_tab);
        rc.x = eda_lookup(c.x, s_tab); rc.y = eda_lookup(c.y, s_tab);
        rc.z = eda_lookup(c.z, s_tab); rc.w = eda_lookup(c.w, s_tab);
        rd.x = eda_lookup(d.x, s_tab); rd.y = eda_lookup(d.y, s_tab);
        rd.z = eda_lookup(d.z, s_tab); rd.w = eda_lookup(d.w, s_tab);
        __builtin_nontemporal_store(ra, y4 + i);
        __builtin_nontemporal_store(rb, y4 + i + stride);
        __builtin_nontemporal_store(rc, y4 + i + 2 * stride);
        __builtin_nontemporal_store(rd, y4 + i + 3 * stride);
    }
    for (; i < n4; i += stride) {
        v4f a = __builtin_nontemporal_load(x4 + i);
        v4f r;
        r.x = eda_lookup(a.x, s_tab);
        r.y = eda_lookup(a.y, s_tab);
        r.z = eda_lookup(a.z, s_tab);
        r.w = eda_lookup(a.w, s_tab);
        __builtin_nontemporal_store(r, y4 + i);
    }

    // Scalar tail for n % 4 (absent for the given 64M shape, kept generic).
    const long long base = n4 << 2;
    for (long long j = base + (long long)blockIdx.x * blockDim.x + threadIdx.x;
         j < n; j += stride) {
        y[j] = eda_lookup(x[j], s_tab);
    }
}

// ---------------------------------------------------------------------------
extern "C" void kernel_launch(void* const* d_in, const int* in_sizes, int n_in,
                              void* d_out, int out_size, void* d_ws, size_t ws_size,
                              hipStream_t stream) {
    const float* x          = (const float*)d_in[0];
    const float* bin_starts = (const float*)d_in[1];
    const float* bin_ends   = (const float*)d_in[2];
    const int*   base_off   = (const int*)d_in[3];
    const int*   k_bits     = (const int*)d_in[4];
    const float* lut_values = (const float*)d_in[5];
    float*       out        = (float*)d_out;
    uint16_t*    table      = (uint16_t*)d_ws;     // 128 KB scratch

    const int  n_bins   = in_sizes[1];
    const int  lut_size = in_sizes[5];
    const long long n   = (long long)in_sizes[0];

    eda_build_table<<<TABLE_ENTRIES / 256, 256, 0, stream>>>(
        bin_starts, bin_ends, base_off, k_bits, lut_values,
        n_bins, lut_size, table);

    const int blocks = 1024;   // 128KB LDS/block -> 2 blocks/WGP resident
    eda_apply<<<blocks, 256, 0, stream>>>(x, out, table, n);
}